// Autoencoder_84911503442556
// MI455X (gfx1250) — compile-verified
//
#include <hip/hip_runtime.h>
#include <math.h>

// ---- problem constants (from reference) ----
#define BH    16          // B*H = 2*8
#define LQC   4096
#define LKC   4096
#define DC    64
#define SKC   45          // SAMPLE_K = 5*ceil(ln 4096) = 45
#define NTC   45          // N_TOP     = 45
#define NTP   48          // padded to 3 tiles of 16
#define SCALEC 0.125f     // 1/sqrt(64)

typedef float v2f __attribute__((ext_vector_type(2)));
typedef float v8f __attribute__((ext_vector_type(8)));

// =====================================================================
// K1: sparsity measure  M[bh][l] = -log( sum_s exp(x_s - max_s) )
//     x_s = q_l . k[index_sample[l][s]]
//     one wave per query; lanes split samples; Q row staged in LDS
// =====================================================================
__global__ __launch_bounds__(256) void k1_sample_scores(
    const float* __restrict__ q, const float* __restrict__ kk,
    const int* __restrict__ idx, float* __restrict__ Mbuf) {
  const int tid  = threadIdx.x;
  const int wave = tid >> 5;
  const int lane = tid & 31;
  const int gq   = blockIdx.x * 8 + wave;   // global query over all bh
  const int bh   = gq >> 12;                // / 4096
  const int l    = gq & 4095;

  __shared__ __align__(16) float sQ[8][64];
  // cooperative row load: 2 floats per lane, coalesced
  *reinterpret_cast<float2*>(&sQ[wave][2 * lane]) =
      *reinterpret_cast<const float2*>(q + ((size_t)bh * LQC + l) * DC + 2 * lane);
  __syncthreads();

  const float* kbh = kk + (size_t)bh * LKC * DC;
  float x0 = -INFINITY, x1 = -INFINITY;
#pragma unroll
  for (int t = 0; t < 2; ++t) {
    int s = lane + 32 * t;
    if (s < SKC) {
      int krow = idx[l * SKC + s];
      const float4* kp = reinterpret_cast<const float4*>(kbh + (size_t)krow * DC);
      float acc = 0.0f;
#pragma unroll
      for (int d4 = 0; d4 < 16; ++d4) {
        float4 kv = kp[d4];
        float4 qv = *reinterpret_cast<const float4*>(&sQ[wave][d4 * 4]);
        acc += qv.x * kv.x + qv.y * kv.y + qv.z * kv.z + qv.w * kv.w;
      }
      if (t == 0) x0 = acc; else x1 = acc;
    }
  }
  float m = fmaxf(x0, x1);
#pragma unroll
  for (int off = 1; off < 32; off <<= 1) m = fmaxf(m, __shfl_xor(m, off, 32));
  float e = ((x0 > -INFINITY) ? __expf(x0 - m) : 0.0f) +
            ((x1 > -INFINITY) ? __expf(x1 - m) : 0.0f);
#pragma unroll
  for (int off = 1; off < 32; off <<= 1) e += __shfl_xor(e, off, 32);
  if (lane == 0) Mbuf[(size_t)bh * LQC + l] = -__logf(e);
}

// =====================================================================
// K2: top-45 selection per (b,h), fully in LDS, lowest-index tie-break
// =====================================================================
__global__ __launch_bounds__(256) void k2_topk(
    const float* __restrict__ Mbuf, int* __restrict__ topk) {
  const int bh  = blockIdx.x;
  const int tid = threadIdx.x;
  __shared__ float sM[LQC];
  __shared__ float rv[256];
  __shared__ int   ri[256];
  for (int i = tid; i < LQC; i += 256) sM[i] = Mbuf[(size_t)bh * LQC + i];
  __syncthreads();
  for (int it = 0; it < NTC; ++it) {
    float bv = -INFINITY; int bi = 0x7fffffff;
    for (int i = tid; i < LQC; i += 256) {
      float vv = sM[i];
      if (vv > bv || (vv == bv && i < bi)) { bv = vv; bi = i; }
    }
    rv[tid] = bv; ri[tid] = bi;
    __syncthreads();
    for (int stride = 128; stride > 0; stride >>= 1) {
      if (tid < stride) {
        float ov = rv[tid + stride]; int oi = ri[tid + stride];
        if (ov > rv[tid] || (ov == rv[tid] && oi < ri[tid])) { rv[tid] = ov; ri[tid] = oi; }
      }
      __syncthreads();
    }
    if (tid == 0) { topk[bh * NTP + it] = ri[0]; sM[ri[0]] = -INFINITY; }
    __syncthreads();
  }
  if (tid == 0) {  // pad rows: allow all keys, never scattered
    topk[bh * NTP + 45] = LQC - 1;
    topk[bh * NTP + 46] = LQC - 1;
    topk[bh * NTP + 47] = LQC - 1;
  }
}

// =====================================================================
// K3: flash-style attention for one 16-query tile of one (b,h).
//     8 waves split the 256 key tiles; V_WMMA_F32_16X16X4_F32 for both
//     S = Q K^T  and  O += P V.  Cross-wave online-softmax merge in LDS.
//
//  fp32 WMMA fragment layout (ISA 7.12.2, 32-bit 16x4 A / 4x16 B):
//     element(row = lane%16, K = 2*(lane/16) + vgpr)
//  so A and B fragments load identically from row-major 16x64 tiles.
// =====================================================================
__global__ __launch_bounds__(256) void k3_attn(
    const float* __restrict__ q, const float* __restrict__ kk,
    const float* __restrict__ vv, const int* __restrict__ topk,
    float* __restrict__ upd) {
  const int bh    = blockIdx.x / 3;
  const int mtile = blockIdx.x % 3;
  const int tid   = threadIdx.x;
  const int wave  = tid >> 5;
  const int lane  = tid & 31;
  const int g     = (lane >> 4) & 1;   // half-wave group
  const int n     = lane & 15;         // column / row index within tile

  __shared__ __align__(16) float sQ[16 * 64];     // gathered Q tile
  __shared__ int   sTop[16];
  __shared__ __align__(8) float sP[8][16 * 18];   // wave-private P (stride 18 keeps b64 aligned)
  __shared__ float sAcc[8][16 * 64];              // per-wave partial O
  __shared__ float sMw[8][16], sLw[8][16];        // per-wave m, l
  __shared__ float sFm[16], sFl[16];

  if (tid < 16) sTop[tid] = topk[bh * NTP + mtile * 16 + tid];
  __syncthreads();
  {  // gather 16 Q rows (16x64 floats), 4 contiguous floats per thread
    int flat = tid * 4;
    int r = flat >> 6, c = flat & 63;
    int row = sTop[r];
    *reinterpret_cast<float4*>(&sQ[r * 64 + c]) =
        *reinterpret_cast<const float4*>(q + ((size_t)bh * LQC + row) * DC + c);
  }
  __syncthreads();

  // Q A-fragments: a[s].{x,y} = Q[lane%16][4s + 2g + {0,1}], s = 0..15 (K=64)
  v2f afr[16];
#pragma unroll
  for (int s = 0; s < 16; ++s)
    afr[s] = *reinterpret_cast<const v2f*>(&sQ[n * 64 + 4 * s + 2 * g]);
  int mt[8];
#pragma unroll
  for (int r = 0; r < 8; ++r) mt[r] = sTop[r + 8 * g];

  v8f acc[4];
#pragma unroll
  for (int nd = 0; nd < 4; ++nd) acc[nd] = v8f{};
  float mrow[8], lrow[8];
#pragma unroll
  for (int r = 0; r < 8; ++r) { mrow[r] = -INFINITY; lrow[r] = 0.0f; }

  const float* kbase = kk + (size_t)bh * LKC * DC;
  const float* vbase = vv + (size_t)bh * LKC * DC;

  for (int kt = wave; kt < LKC / 16; kt += 8) {
    const int kb = kt * 16;
    __builtin_prefetch(kbase + (size_t)(kb + 128 + n) * DC, 0, 1);  // next tile for this wave

    // ---- S = Q K^T via 16 chained fp32 WMMAs (K = 64) ----
    v8f S = v8f{};
#pragma unroll
    for (int s = 0; s < 16; ++s) {
      v2f bfr = *reinterpret_cast<const v2f*>(kbase + (size_t)(kb + n) * DC + 4 * s + 2 * g);
      S = __builtin_amdgcn_wmma_f32_16x16x4_f32(false, afr[s], false, bfr,
                                                (short)0, S, false, false);
    }

    // ---- causal mask + per-row online softmax (rows m = r + 8g, cols = lanes) ----
    float p[8];
#pragma unroll
    for (int r = 0; r < 8; ++r) {
      float sc = S[r] * SCALEC;
      if (kb + n > mt[r]) sc = -INFINITY;
      float tmax = sc;
#pragma unroll
      for (int off = 1; off < 16; off <<= 1)
        tmax = fmaxf(tmax, __shfl_xor(tmax, off, 16));
      float mnew = fmaxf(mrow[r], tmax);
      float pr   = (sc > -INFINITY) ? __expf(sc - mnew) : 0.0f;
      float tsum = pr;
#pragma unroll
      for (int off = 1; off < 16; off <<= 1) tsum += __shfl_xor(tsum, off, 16);
      float alpha = (mrow[r] > -INFINITY) ? __expf(mrow[r] - mnew) : 0.0f;
      lrow[r] = lrow[r] * alpha + tsum;
      mrow[r] = mnew;
#pragma unroll
      for (int nd = 0; nd < 4; ++nd) acc[nd][r] *= alpha;
      p[r] = pr;
    }

    // ---- C-layout -> A-layout for P via wave-private LDS tile ----
#pragma unroll
    for (int r = 0; r < 8; ++r) sP[wave][(r + 8 * g) * 18 + n] = p[r];
    asm volatile("s_wait_dscnt 0x0" ::: "memory");  // LDS stores visible to our loads
    v2f pa[4];
#pragma unroll
    for (int s = 0; s < 4; ++s)
      pa[s] = *reinterpret_cast<const v2f*>(&sP[wave][n * 18 + 4 * s + 2 * g]);

    // ---- O += P V via 16 fp32 WMMAs (4 dim-tiles x K=16) ----
#pragma unroll
    for (int nd = 0; nd < 4; ++nd) {
#pragma unroll
      for (int s = 0; s < 4; ++s) {
        v2f bv;
        bv.x = vbase[(size_t)(kb + 4 * s + 2 * g + 0) * DC + nd * 16 + n];
        bv.y = vbase[(size_t)(kb + 4 * s + 2 * g + 1) * DC + nd * 16 + n];
        acc[nd] = __builtin_amdgcn_wmma_f32_16x16x4_f32(false, pa[s], false, bv,
                                                        (short)0, acc[nd], false, false);
      }
    }
  }

  // ---- publish per-wave partials ----
  if (n == 0) {
#pragma unroll
    for (int r = 0; r < 8; ++r) {
      sMw[wave][r + 8 * g] = mrow[r];
      sLw[wave][r + 8 * g] = lrow[r];
    }
  }
#pragma unroll
  for (int nd = 0; nd < 4; ++nd)
#pragma unroll
    for (int r = 0; r < 8; ++r)
      sAcc[wave][(r + 8 * g) * 64 + nd * 16 + n] = acc[nd][r];
  __syncthreads();

  // ---- cross-wave softmax merge ----
  if (tid < 16) {
    float gm = -INFINITY;
    for (int w = 0; w < 8; ++w) gm = fmaxf(gm, sMw[w][tid]);
    float gl = 0.0f;
    for (int w = 0; w < 8; ++w)
      if (sMw[w][tid] > -INFINITY) gl += sLw[w][tid] * __expf(sMw[w][tid] - gm);
    sFm[tid] = gm; sFl[tid] = gl;
  }
  __syncthreads();
  {
    int m  = tid >> 4;
    int d0 = (tid & 15) * 4;
    float gm = sFm[m], gl = sFl[m];
    float4 o; float* op = &o.x;
#pragma unroll
    for (int i = 0; i < 4; ++i) {
      float s = 0.0f;
      for (int w = 0; w < 8; ++w) {
        float wm = sMw[w][m];
        if (wm > -INFINITY) s += sAcc[w][m * 64 + d0 + i] * __expf(wm - gm);
      }
      op[i] = s / gl;
    }
    *reinterpret_cast<float4*>(upd + ((size_t)bh * NTP + mtile * 16 + m) * DC + d0) = o;
  }
}

// =====================================================================
// K4: context = cumsum(V, axis=2).  Two-pass coalesced scan:
//     1024 threads = 16 chunk-groups x 64 dims per (b,h)
// =====================================================================
__global__ __launch_bounds__(1024) void k4_cumsum(
    const float* __restrict__ vv, float* __restrict__ out) {
  const int bh  = blockIdx.x;
  const int tid = threadIdx.x;
  const int d   = tid & 63;
  const int c   = tid >> 6;  // chunk 0..15, 256 rows each
  __shared__ float part[16][64];
  const float* vb = vv + (size_t)bh * LKC * DC;
  float* ob = out + (size_t)bh * LKC * DC;
  const int r0 = c * 256;
  float s = 0.0f;
  for (int j = 0; j < 256; ++j) s += vb[(size_t)(r0 + j) * DC + d];
  part[c][d] = s;
  __syncthreads();
  float run = 0.0f;
  for (int cc = 0; cc < 16; ++cc) if (cc < c) run += part[cc][d];
  for (int j = 0; j < 256; ++j) {
    run += vb[(size_t)(r0 + j) * DC + d];
    ob[(size_t)(r0 + j) * DC + d] = run;
  }
}

// =====================================================================
// K5: scatter the 45 attention rows into the cumsum context
// =====================================================================
__global__ __launch_bounds__(256) void k5_scatter(
    const float* __restrict__ upd, const int* __restrict__ topk,
    float* __restrict__ out) {
  const int bh  = blockIdx.x;
  const int tid = threadIdx.x;
  for (int e = tid; e < NTC * DC; e += 256) {
    int u = e >> 6, dd = e & 63;
    int row = topk[bh * NTP + u];
    out[((size_t)bh * LQC + row) * DC + dd] = upd[((size_t)bh * NTP + u) * DC + dd];
  }
}

// =====================================================================
extern "C" void kernel_launch(void* const* d_in, const int* in_sizes, int n_in,
                              void* d_out, int out_size, void* d_ws, size_t ws_size,
                              hipStream_t stream) {
  (void)in_sizes; (void)n_in; (void)out_size; (void)ws_size;
  const float* q  = (const float*)d_in[0];
  const float* k  = (const float*)d_in[1];
  const float* v  = (const float*)d_in[2];
  const int* idx  = (const int*)d_in[3];
  float* out      = (float*)d_out;

  float* Mbuf = (float*)d_ws;                         // BH*LQ floats   (256 KB)
  int*   topk = (int*)(Mbuf + (size_t)BH * LQC);      // BH*NTP ints    (3 KB)
  float* upd  = (float*)(topk + BH * NTP);            // BH*NTP*D f32   (192 KB), 16B aligned

  k1_sample_scores<<<BH * LQC / 8, 256, 0, stream>>>(q, k, idx, Mbuf);
  k2_topk<<<BH, 256, 0, stream>>>(Mbuf, topk);
  k3_attn<<<BH * 3, 256, 0, stream>>>(q, k, v, topk, upd);
  k4_cumsum<<<BH, 1024, 0, stream>>>(v, out);
  k5_scatter<<<BH, 256, 0, stream>>>(upd, topk, out);
}